// TsCv_29257317220818
// MI455X (gfx1250) — compile-verified
//
#include <hip/hip_runtime.h>
#include <hip/hip_bf16.h>
#include <math.h>

#define WIN     20
#define WSTRIDE 5
#define EPS_F   1e-8f
#define S_LEN   4096
#define OUT_LEN 816          // (4096 - 20)/5 + 1
#define NPAIR   408          // OUT_LEN / 2 : two consecutive windows per thread
#define BLOCK   256

// b128 async-copy operand types: int4 pointers in global (AS1) / LDS (AS3).
typedef int v4i __attribute__((vector_size(16)));
typedef __attribute__((address_space(1))) v4i* gbl_v4i_p;
typedef __attribute__((address_space(3))) v4i* lds_v4i_p;

__global__ __launch_bounds__(BLOCK)
void TsCv_29257317220818_kernel(const float* __restrict__ x,
                                float* __restrict__ out) {
    __shared__ float row[S_LEN];

    const int tid = threadIdx.x;
    const long long r = blockIdx.x;                       // row index (b*F + f)
    const float* __restrict__ gsrc = x + r * (long long)S_LEN;

    // ---- Stage one 16KB row into LDS via async DMA --------------------
#if __has_builtin(__builtin_amdgcn_global_load_async_to_lds_b128)
    {
        // 256 lanes * 16B = 4KB per iteration; 4 iterations = 16KB row.
        #pragma unroll
        for (int it = 0; it < 4; ++it) {
            const int vec_idx = tid + it * BLOCK;         // int4 index
            __builtin_amdgcn_global_load_async_to_lds_b128(
                (gbl_v4i_p)(gsrc + 4 * vec_idx),
                (lds_v4i_p)(&row[4 * vec_idx]),
                /*imm offset*/ 0, /*cpol*/ 0);
        }
#if __has_builtin(__builtin_amdgcn_s_wait_asynccnt)
        __builtin_amdgcn_s_wait_asynccnt(0);
#else
        asm volatile("s_wait_asynccnt 0" ::: "memory");
#endif
    }
#else
    {
        // Fallback: vectorized load through VGPRs.
        const float4* __restrict__ s4 = (const float4*)gsrc;
        float4* d4 = (float4*)row;
        #pragma unroll
        for (int it = 0; it < 4; ++it) {
            const int i = tid + it * BLOCK;
            d4[i] = s4[i];
        }
    }
#endif
    __syncthreads();

    // ---- Compute CV for window pairs from LDS -------------------------
    // Pair p = windows {2p, 2p+1}: floats [10p, 10p+25). Per-lane LDS
    // address stride = 10 floats -> conflict-free across a wave32.
    float2* __restrict__ out2 = (float2*)(out + r * (long long)OUT_LEN);

    #pragma unroll
    for (int k = 0; k < 2; ++k) {
        const int p = tid + k * BLOCK;
        if (p < NPAIR) {
            const int start = p * (2 * WSTRIDE);
            float v[WIN + WSTRIDE];                       // 25 floats
            #pragma unroll
            for (int j = 0; j < WIN + WSTRIDE; ++j) v[j] = row[start + j];

            float sumA = 0.0f;
            #pragma unroll
            for (int j = 0; j < WIN; ++j) sumA += v[j];
            float head5 = 0.0f, tail5 = 0.0f;
            #pragma unroll
            for (int j = 0; j < WSTRIDE; ++j) {
                head5 += v[j];
                tail5 += v[WIN + j];
            }
            const float sumB  = sumA - head5 + tail5;     // window B sum (reuse)
            const float meanA = sumA * (1.0f / (float)WIN);
            const float meanB = sumB * (1.0f / (float)WIN);

            float accA = 0.0f, accB = 0.0f;
            #pragma unroll
            for (int j = 0; j < WIN; ++j) {
                const float dA = v[j] - meanA;
                accA = fmaf(dA, dA, accA);
                const float dB = v[WSTRIDE + j] - meanB;
                accB = fmaf(dB, dB, accB);
            }
            float cvA = sqrtf(accA * (1.0f / (float)(WIN - 1))) / (meanA + EPS_F);
            float cvB = sqrtf(accB * (1.0f / (float)(WIN - 1))) / (meanB + EPS_F);
            if (cvA != cvA) cvA = 0.0f;                   // NaN -> 0
            if (cvB != cvB) cvB = 0.0f;

            float2 o; o.x = cvA; o.y = cvB;
            out2[p] = o;                                  // aligned b64 store
        }
    }
}

extern "C" void kernel_launch(void* const* d_in, const int* in_sizes, int n_in,
                              void* d_out, int out_size, void* d_ws, size_t ws_size,
                              hipStream_t stream) {
    (void)n_in; (void)out_size; (void)d_ws; (void)ws_size;
    const float* x = (const float*)d_in[0];
    float* out = (float*)d_out;
    const int rows = in_sizes[0] / S_LEN;                 // 64*256 = 16384
    TsCv_29257317220818_kernel<<<rows, BLOCK, 0, stream>>>(x, out);
}